// ODEFunction_37194416783837
// MI455X (gfx1250) — compile-verified
//
#include <hip/hip_runtime.h>

// out[row, :] += val * x[col, :]  for each COO edge (row, col, val)
// N_NODES=100000, N_EDGES=3200000, D_FEAT=64, all f32.
//
// Memory-bound scatter: x/out (25.6MB each) stay resident in the 192MB L2;
// HBM traffic is the streamed edge lists (~38MB). Bottleneck = L2 f32-atomic
// rate: 1 wave per edge, 2 wave-wide non-returning global_atomic_add_f32 per
// edge (the minimum for f32 data), edge metadata on the scalar path via
// s_load_b128 chunks, 4-edge software pipeline with gfx1250 global_prefetch
// on the next chunk's x rows. All offsets kept in unsigned 32-bit so the
// backend uses SADDR-base + 32-bit-voffset global addressing (no 64-bit
// VALU address math, fewer VGPRs -> more waves resident).

#define D_FEAT 64u
#define EDGES_PER_WAVE 128
#define WAVES_PER_BLOCK 8  // 256 threads, wave32

__global__ __launch_bounds__(256) void gcn_spmm_coo_kernel(
    const float* __restrict__ x,
    const int*   __restrict__ edge_row,
    const int*   __restrict__ edge_col,
    const float* __restrict__ edge_val,
    float*       __restrict__ out,
    int n_edges)
{
    const unsigned lane2 = (threadIdx.x & 31u) * 2u;  // 2 floats per lane
    // Wave-uniform edge chunk; readfirstlane forces SGPR so all edge_* loads
    // stay on the scalar (SMEM) path.
    const int wave = __builtin_amdgcn_readfirstlane(
        blockIdx.x * WAVES_PER_BLOCK + (threadIdx.x >> 5));

    const int e0 = wave * EDGES_PER_WAVE;
    if (e0 >= n_edges) return;
    int e1 = e0 + EDGES_PER_WAVE;
    if (e1 > n_edges) e1 = n_edges;

    int e = e0;
    if (e + 4 <= e1) {
        // Pipeline: current chunk's cols already in SGPRs; fetch next chunk's
        // cols early and prefetch their x rows while processing this chunk.
        int4 col4 = *(const int4*)(edge_col + e);  // s_load_b128
        for (; e + 4 <= e1; e += 4) {
            const int4   row4 = *(const int4*)(edge_row + e);
            const float4 val4 = *(const float4*)(edge_val + e);

            // 32-bit element offsets (max 6.4M floats -> fits easily)
            const unsigned c0 = (unsigned)col4.x * D_FEAT + lane2;
            const unsigned c1 = (unsigned)col4.y * D_FEAT + lane2;
            const unsigned c2 = (unsigned)col4.z * D_FEAT + lane2;
            const unsigned c3 = (unsigned)col4.w * D_FEAT + lane2;

            if (e + 8 <= e1) {
                col4 = *(const int4*)(edge_col + e + 4);
                // 32 lanes x 8B = the full 256B row per prefetch
                __builtin_prefetch(x + ((unsigned)col4.x * D_FEAT + lane2), 0, 0);
                __builtin_prefetch(x + ((unsigned)col4.y * D_FEAT + lane2), 0, 0);
                __builtin_prefetch(x + ((unsigned)col4.z * D_FEAT + lane2), 0, 0);
                __builtin_prefetch(x + ((unsigned)col4.w * D_FEAT + lane2), 0, 0);
            }

            // 4 independent gathers in flight (global_load_b64 each)
            const float2 g0 = *(const float2*)(x + c0);
            const float2 g1 = *(const float2*)(x + c1);
            const float2 g2 = *(const float2*)(x + c2);
            const float2 g3 = *(const float2*)(x + c3);

            const unsigned r0 = (unsigned)row4.x * D_FEAT + lane2;
            const unsigned r1 = (unsigned)row4.y * D_FEAT + lane2;
            const unsigned r2 = (unsigned)row4.z * D_FEAT + lane2;
            const unsigned r3 = (unsigned)row4.w * D_FEAT + lane2;

            // 8 wave-wide non-returning f32 atomics (2 per edge, the minimum)
            unsafeAtomicAdd(out + r0,     g0.x * val4.x);
            unsafeAtomicAdd(out + r0 + 1, g0.y * val4.x);
            unsafeAtomicAdd(out + r1,     g1.x * val4.y);
            unsafeAtomicAdd(out + r1 + 1, g1.y * val4.y);
            unsafeAtomicAdd(out + r2,     g2.x * val4.z);
            unsafeAtomicAdd(out + r2 + 1, g2.y * val4.z);
            unsafeAtomicAdd(out + r3,     g3.x * val4.w);
            unsafeAtomicAdd(out + r3 + 1, g3.y * val4.w);
        }
    }
    // Scalar tail (only the last wave's chunk, only if n_edges % 4 != 0)
    for (; e < e1; ++e) {
        const unsigned col = (unsigned)edge_col[e] * D_FEAT + lane2;
        const unsigned row = (unsigned)edge_row[e] * D_FEAT + lane2;
        const float v = edge_val[e];
        const float2 g = *(const float2*)(x + col);
        unsafeAtomicAdd(out + row,     g.x * v);
        unsafeAtomicAdd(out + row + 1, g.y * v);
    }
}

__global__ __launch_bounds__(256) void gcn_zero_kernel(float4* __restrict__ out, int n4) {
    int i = blockIdx.x * 256 + threadIdx.x;
    if (i < n4) out[i] = make_float4(0.f, 0.f, 0.f, 0.f);
}

extern "C" void kernel_launch(void* const* d_in, const int* in_sizes, int n_in,
                              void* d_out, int out_size, void* d_ws, size_t ws_size,
                              hipStream_t stream) {
    // d_in: 0=t (unused), 1=x [N_NODES*64 f32], 2=edge_row, 3=edge_col, 4=edge_val
    const float* x        = (const float*)d_in[1];
    const int*   edge_row = (const int*)  d_in[2];
    const int*   edge_col = (const int*)  d_in[3];
    const float* edge_val = (const float*)d_in[4];
    float*       out      = (float*)d_out;

    const int n_edges = in_sizes[2];

    // 1) zero the output (harness poisons it), float4 stores
    const int n4 = out_size / 4;
    gcn_zero_kernel<<<(n4 + 255) / 256, 256, 0, stream>>>((float4*)out, n4);

    // 2) edge scatter
    const int edges_per_block = EDGES_PER_WAVE * WAVES_PER_BLOCK;
    const int nblocks = (n_edges + edges_per_block - 1) / edges_per_block;
    gcn_spmm_coo_kernel<<<nblocks, 256, 0, stream>>>(
        x, edge_row, edge_col, edge_val, out, n_edges);
}